// SNN_42545946034957
// MI455X (gfx1250) — compile-verified
//
#include <hip/hip_runtime.h>

typedef float v2f __attribute__((ext_vector_type(2)));
typedef float v8f __attribute__((ext_vector_type(8)));

// Problem constants (fixed by the reference setup)
constexpr int   D_IN = 187;   // input features
constexpr int   H1   = 50;    // hidden neurons
constexpr int   NC   = 5;     // output classes
constexpr float BETA = 0.9f;
constexpr float THR  = 1.0f;

// ---------------------------------------------------------------------------
// Phase 1: cur1 = x @ W1^T + b1 via V_WMMA_F32_16X16X4_F32.
// One wave (32 threads) per 16-row tile of x; 4 accumulators cover output
// columns 0..63 (only 0..49 stored). K marches in steps of 4: 46 full steps
// plus a 3-wide tail.
//
// Guard-free inner loop:
//  * B rows for n >= 50 are CLAMPED to row 49 (in-bounds load). A B-column
//    only feeds the same D-column, and columns 50..63 are never stored, so
//    the duplicated data is harmless garbage.
//  * K-tail: only a.y of the upper half-wave (k = 187) is out of range.
//    Zeroing a.y makes every product with B row k=187 zero, so B tail loads
//    just use a clamped index — no exec-mask branches anywhere.
//
// Fragment layouts (ISA 7.12.2, 32-bit 16x4 A / 4x16 B / 16x16 C):
//  A: lane&15 = M row, (lane>>4)*2 = K pair; v0=A[m][kb], v1=A[m][kb+1]
//  B: lane&15 = N col, same half-wave K-pair split
//  C/D: VGPR i holds row m0 + 8*(lane>>4) + i, col = lane&15
// ---------------------------------------------------------------------------
__global__ __launch_bounds__(32)
void snn_fc1_wmma(const float* __restrict__ x,
                  const float* __restrict__ W1,
                  const float* __restrict__ b1,
                  float* __restrict__ cur1)
{
    const int lane = threadIdx.x & 31;
    const int m0   = blockIdx.x * 16;
    const int mrow = m0 + (lane & 15);
    const int kb   = (lane >> 4) * 2;      // 0 or 2
    const int ncol = lane & 15;

    const float* __restrict__ xrow = x + (size_t)mrow * D_IN;

    // Per-lane clamped B row pointers (unconditional in-bounds loads)
    const float* __restrict__ wrow[4];
#pragma unroll
    for (int nt = 0; nt < 4; ++nt) {
        int n = nt * 16 + ncol;
        if (n > H1 - 1) n = H1 - 1;        // clamp: dup of col 49, never stored
        wrow[nt] = W1 + (size_t)n * D_IN;
    }

    v8f acc[4];
#pragma unroll
    for (int nt = 0; nt < 4; ++nt) acc[nt] = (v8f){};

    int k0 = 0;
#pragma unroll 1
    for (; k0 + 4 <= D_IN; k0 += 4) {       // 46 full K-steps, branch-free body
        v2f a;
        a.x = xrow[k0 + kb];
        a.y = xrow[k0 + kb + 1];
#pragma unroll
        for (int nt = 0; nt < 4; ++nt) {
            v2f b;
            b.x = wrow[nt][k0 + kb];
            b.y = wrow[nt][k0 + kb + 1];
            acc[nt] = __builtin_amdgcn_wmma_f32_16x16x4_f32(
                false, a, false, b, (short)0, acc[nt], false, false);
        }
    }

    { // K tail: k0 == 184, valid k in {184,185,186}
        const int  kk   = k0 + kb;                     // 184 or 186, in range
        const bool y_ok = (kk + 1 < D_IN);             // false only for kb==2
        const int  kk1  = y_ok ? (kk + 1) : (D_IN - 1); // clamped index

        v2f a;
        a.x = xrow[kk];
        a.y = y_ok ? xrow[kk1] : 0.0f;                 // zero kills k=187 term
#pragma unroll
        for (int nt = 0; nt < 4; ++nt) {
            v2f b;
            b.x = wrow[nt][kk];
            b.y = wrow[nt][kk1];                       // value irrelevant if !y_ok
            acc[nt] = __builtin_amdgcn_wmma_f32_16x16x4_f32(
                false, a, false, b, (short)0, acc[nt], false, false);
        }
    }

    // Epilogue: add bias, store the valid 16x50 tile of cur1
    const int rbase = m0 + 8 * (lane >> 4);
#pragma unroll
    for (int nt = 0; nt < 4; ++nt) {
        const int col = nt * 16 + ncol;
        if (col < H1) {
            const float bias = b1[col];
#pragma unroll
            for (int i = 0; i < 8; ++i)
                cur1[(size_t)(rbase + i) * H1 + col] = acc[nt][i] + bias;
        }
    }
}

// ---------------------------------------------------------------------------
// Phase 2: 100-step LIF recurrence. One batch row per thread; all state in
// VGPRs; W2/b2 broadcast from LDS. Stores are contiguous 5-float runs per
// thread -> coalesced across the block. Output-store bound (~131 MB).
// ---------------------------------------------------------------------------
__global__ __launch_bounds__(256)
void snn_steps(const float* __restrict__ cur1,
               const float* __restrict__ W2,
               const float* __restrict__ b2,
               float* __restrict__ spk_rec,
               float* __restrict__ mem_rec,
               int Btot, int nsteps)
{
    __shared__ float sW2[NC * H1];
    __shared__ float sb2[NC];
    for (int i = threadIdx.x; i < NC * H1; i += blockDim.x) sW2[i] = W2[i];
    if (threadIdx.x < NC) sb2[threadIdx.x] = b2[threadIdx.x];
    __syncthreads();

    const int b = blockIdx.x * blockDim.x + threadIdx.x;
    if (b >= Btot) return;

    float cur[H1];
    const float* __restrict__ crow = cur1 + (size_t)b * H1;
#pragma unroll
    for (int h = 0; h < H1; ++h) cur[h] = crow[h];

    float mem1[H1];
#pragma unroll
    for (int h = 0; h < H1; ++h) mem1[h] = 0.0f;
    float mem2[NC];
#pragma unroll
    for (int c = 0; c < NC; ++c) mem2[c] = 0.0f;

#pragma unroll 1
    for (int s = 0; s < nsteps; ++s) {
        float cur2[NC];
#pragma unroll
        for (int c = 0; c < NC; ++c) cur2[c] = sb2[c];

#pragma unroll
        for (int h = 0; h < H1; ++h) {
            // reset from PREVIOUS mem (stop_gradient path), subtract mechanism
            const float r = (mem1[h] > THR) ? THR : 0.0f;
            mem1[h] = fmaf(BETA, mem1[h], cur[h] - r);
            // spike = Heaviside(mem - THR)
            const float spk = (mem1[h] > THR) ? 1.0f : 0.0f;
#pragma unroll
            for (int c = 0; c < NC; ++c)
                cur2[c] = fmaf(spk, sW2[c * H1 + h], cur2[c]);
        }

        const size_t base = ((size_t)s * Btot + b) * NC;
#pragma unroll
        for (int c = 0; c < NC; ++c) {
            const float r2 = (mem2[c] > THR) ? THR : 0.0f;
            mem2[c] = fmaf(BETA, mem2[c], cur2[c] - r2);
            const float spk2 = (mem2[c] > THR) ? 1.0f : 0.0f;
            spk_rec[base + c] = spk2;
            mem_rec[base + c] = mem2[c];
        }
    }
}

// ---------------------------------------------------------------------------
extern "C" void kernel_launch(void* const* d_in, const int* in_sizes, int n_in,
                              void* d_out, int out_size, void* d_ws, size_t ws_size,
                              hipStream_t stream)
{
    const float* x  = (const float*)d_in[0];
    const float* W1 = (const float*)d_in[1];
    const float* b1 = (const float*)d_in[2];
    const float* W2 = (const float*)d_in[3];
    const float* b2 = (const float*)d_in[4];

    // in_sizes: x=B*D, W1=H*D, b1=H, W2=C*H, b2=C, num_steps(scalar)=1
    const int H = in_sizes[2];                 // 50
    const int C = in_sizes[4];                 // 5
    const int D = in_sizes[1] / H;             // 187
    const int B = in_sizes[0] / D;             // 32768
    const int S = out_size / (2 * B * C);      // 100 (out = spk ++ mem)

    float* cur1    = (float*)d_ws;                          // B*H f32 scratch
    float* spk_rec = (float*)d_out;
    float* mem_rec = (float*)d_out + (size_t)S * B * C;

    snn_fc1_wmma<<<dim3((B + 15) / 16), dim3(32), 0, stream>>>(x, W1, b1, cur1);
    snn_steps<<<dim3((B + 255) / 256), dim3(256), 0, stream>>>(
        cur1, W2, b2, spk_rec, mem_rec, B, S);
}